// GNN_62294205661279
// MI455X (gfx1250) — compile-verified
//
#include <hip/hip_runtime.h>
#include <hip/hip_bf16.h>

typedef __attribute__((ext_vector_type(2))) float v2f;
typedef __attribute__((ext_vector_type(8))) float v8f;

#define BN_EPS 1e-5f

// ---------------- degree / normalization ----------------

__global__ void k_init_deg(float* __restrict__ deg, long long N) {
  long long i = blockIdx.x * 256LL + threadIdx.x;
  if (i < N) deg[i] = 1.0f;  // self-loop contribution
}

__global__ void k_deg_edges(const long long* __restrict__ ei, float* __restrict__ deg,
                            long long E) {
  long long i = blockIdx.x * 256LL + threadIdx.x;
  if (i < E) atomicAdd(&deg[ei[E + i]], 1.0f);  // dst row of edge_index[2,E]
}

__global__ void k_deg_to_dinv(float* __restrict__ deg, long long N) {
  long long i = blockIdx.x * 256LL + threadIdx.x;
  if (i < N) deg[i] = rsqrtf(deg[i]);  // deg >= 1 always (self loop)
}

// ---------------- dense transform: H = X @ W (NxK @ KxK, K=128), fp32 WMMA ----------------
// Block: 256 threads = 8 waves. Block tile: 128 rows x 128 cols.
// Wave w owns column strip [16w, 16w+16) and iterates 8 row tiles of 16.
// LDS stride 132 floats (mod 64 banks = 4) => conflict-free b64 fragment loads.

__global__ __launch_bounds__(256) void k_gcn_gemm128(const float* __restrict__ X,
                                                     const float* __restrict__ W,
                                                     float* __restrict__ H, long long N) {
  __shared__ float Xs[128 * 132];
  __shared__ float Wt[128 * 132];
  const int tid = threadIdx.x;
  const long long rowbase = (long long)blockIdx.x * 128;

  // Stage W transposed: Wt[n][k] = W[k][n]  (coalesced global reads)
  for (int idx = tid; idx < 128 * 128; idx += 256) {
    int k = idx >> 7, n = idx & 127;
    Wt[n * 132 + k] = W[idx];
  }
  // Stage X tile row-major with float4, clamp OOB rows (stores are guarded later)
  for (int idx = tid; idx < 128 * 32; idx += 256) {
    int r = idx >> 5, c4 = (idx & 31) << 2;
    long long row = rowbase + r;
    if (row >= N) row = N - 1;
    *(float4*)&Xs[r * 132 + c4] = *(const float4*)&X[row * 128 + c4];
  }
  __syncthreads();

  const int wave = tid >> 5, lane = tid & 31;
  const int n0 = wave << 4;
  const int lm = lane & 15;
  const int kb = (lane >> 4) << 1;  // A/B 16x4 fp32 layout: lanes 16-31 hold K=2,3

  v8f acc[8] = {};
  for (int k0 = 0; k0 < 128; k0 += 4) {
    v2f b = *(const v2f*)&Wt[(n0 + lm) * 132 + k0 + kb];
#pragma unroll
    for (int t = 0; t < 8; ++t) {
      v2f a = *(const v2f*)&Xs[((t << 4) + lm) * 132 + k0 + kb];
      acc[t] = __builtin_amdgcn_wmma_f32_16x16x4_f32(false, a, false, b, (short)0,
                                                     acc[t], false, false);
    }
  }
  // C/D layout: VGPR v holds M = v + 8*(lane>=16), N = lane&15
#pragma unroll
  for (int t = 0; t < 8; ++t) {
#pragma unroll
    for (int v = 0; v < 8; ++v) {
      long long row = rowbase + (t << 4) + v + ((lane >> 4) << 3);
      if (row < N) H[row * 128 + n0 + lm] = acc[t][v];
    }
  }
}

// ---------------- edge aggregation: agg[dst] += h[src] * dinv[src]*dinv[dst] ----------------
// One 32-lane wave per edge; each lane handles 4 contiguous features (float4 gather,
// 4 global_atomic_add_f32). Feature matrix (51 MB) is L2-resident on a 192 MB L2.

__global__ void k_edge_agg(const float* __restrict__ h, const long long* __restrict__ ei,
                           const float* __restrict__ dinv, float* __restrict__ agg,
                           long long E, long long N) {
  long long t = blockIdx.x * 256LL + threadIdx.x;
  if (t >= (E + N) * 32) return;
  long long e = t >> 5;
  int c = (int)(t & 31) << 2;
  long long s, d;
  if (e < E) { s = ei[e]; d = ei[E + e]; }
  else       { s = e - E; d = s; }       // self loop
  float nrm = dinv[s] * dinv[d];
  float4 v = *(const float4*)&h[s * 128 + c];
  float* dst = &agg[d * 128 + c];
  atomicAdd(dst + 0, v.x * nrm);
  atomicAdd(dst + 1, v.y * nrm);
  atomicAdd(dst + 2, v.z * nrm);
  atomicAdd(dst + 3, v.w * nrm);
}

// ---------------- batchnorm ----------------

__global__ __launch_bounds__(256) void k_bn_stats(const float* __restrict__ h,
                                                  float* __restrict__ stats,  // [sum128|sumsq128]
                                                  long long N) {
  __shared__ float s1[256], s2[256];
  int tid = threadIdx.x;
  int col = tid & 127;
  int half = tid >> 7;
  float sum = 0.f, sq = 0.f;
  for (long long row = blockIdx.x * 2LL + half; row < N; row += gridDim.x * 2LL) {
    float v = h[row * 128 + col];
    sum += v; sq += v * v;
  }
  s1[tid] = sum; s2[tid] = sq;
  __syncthreads();
  if (tid < 128) {
    atomicAdd(&stats[col], s1[tid] + s1[tid + 128]);
    atomicAdd(&stats[128 + col], s2[tid] + s2[tid + 128]);
  }
}

__global__ void k_bn_apply(float* __restrict__ h, const float* __restrict__ stats,
                           const float* __restrict__ g, const float* __restrict__ be,
                           long long N) {
  long long i = blockIdx.x * 256LL + threadIdx.x;  // float4 index
  if (i >= N * 32) return;
  int c4 = (int)(i & 31) << 2;
  float4 v = ((float4*)h)[i];
  float invN = 1.0f / (float)N;
  float* vp = &v.x;
#pragma unroll
  for (int j = 0; j < 4; ++j) {
    int c = c4 + j;
    float mu = stats[c] * invN;
    float var = stats[128 + c] * invN - mu * mu;  // biased var, matches jnp var
    float y = (vp[j] - mu) * rsqrtf(var + BN_EPS) * g[c] + be[c];
    vp[j] = fmaxf(y, 0.0f);
  }
  ((float4*)h)[i] = v;
}

// ---------------- global mean pool ----------------

__global__ void k_cnt(const long long* __restrict__ batch, float* __restrict__ cnt,
                      long long N) {
  long long i = blockIdx.x * 256LL + threadIdx.x;
  if (i < N) atomicAdd(&cnt[batch[i]], 1.0f);
}

__global__ void k_pool(const float* __restrict__ h, const long long* __restrict__ batch,
                       float* __restrict__ pooled, long long N) {
  long long t = blockIdx.x * 256LL + threadIdx.x;
  if (t >= N * 32) return;
  long long node = t >> 5;
  int c = (int)(t & 31) << 2;
  long long b = batch[node];
  float4 v = *(const float4*)&h[node * 128 + c];
  float* dst = &pooled[b * 128 + c];
  atomicAdd(dst + 0, v.x);
  atomicAdd(dst + 1, v.y);
  atomicAdd(dst + 2, v.z);
  atomicAdd(dst + 3, v.w);
}

// ---------------- final FC: out[64,32] = (pooled/cnt) @ Wfc + bfc, single block WMMA ----------------

__global__ __launch_bounds__(256) void k_pool_fc(const float* __restrict__ pooled,
                                                 const float* __restrict__ cnt,
                                                 const float* __restrict__ Wfc,
                                                 const float* __restrict__ bfc,
                                                 float* __restrict__ out) {
  __shared__ float Ps[64 * 132];  // normalized pooled, row-major
  __shared__ float Ft[32 * 132];  // Wfc transposed: Ft[n][k]
  int tid = threadIdx.x;
  for (int idx = tid; idx < 64 * 128; idx += 256) {
    int gidx = idx >> 7, c = idx & 127;
    Ps[gidx * 132 + c] = pooled[idx] / fmaxf(cnt[gidx], 1.0f);
  }
  for (int idx = tid; idx < 128 * 32; idx += 256) {
    int k = idx >> 5, n = idx & 31;
    Ft[n * 132 + k] = Wfc[idx];
  }
  __syncthreads();

  int wave = tid >> 5, lane = tid & 31;
  int m0 = (wave >> 1) << 4;  // 4 row tiles (64 graphs)
  int n0 = (wave & 1) << 4;   // 2 col tiles (32 outputs)
  int lm = lane & 15, kb = (lane >> 4) << 1;

  v8f acc = {};
  for (int k0 = 0; k0 < 128; k0 += 4) {
    v2f a = *(const v2f*)&Ps[(m0 + lm) * 132 + k0 + kb];
    v2f b = *(const v2f*)&Ft[(n0 + lm) * 132 + k0 + kb];
    acc = __builtin_amdgcn_wmma_f32_16x16x4_f32(false, a, false, b, (short)0,
                                                acc, false, false);
  }
#pragma unroll
  for (int v = 0; v < 8; ++v) {
    int row = m0 + v + ((lane >> 4) << 3);
    int col = n0 + lm;
    out[row * 32 + col] = acc[v] + bfc[col];
  }
}

// ---------------- host ----------------

extern "C" void kernel_launch(void* const* d_in, const int* in_sizes, int n_in,
                              void* d_out, int out_size, void* d_ws, size_t ws_size,
                              hipStream_t stream) {
  const float*     x     = (const float*)d_in[0];
  const long long* ei    = (const long long*)d_in[1];   // int64 [2,E]
  const long long* batch = (const long long*)d_in[2];   // int64 [N]
  const float* W0  = (const float*)d_in[3];
  // b0 = d_in[4], b1 = d_in[8]: constants per column, cancel exactly under BatchNorm.
  const float* g0  = (const float*)d_in[5];
  const float* be0 = (const float*)d_in[6];
  const float* W1  = (const float*)d_in[7];
  const float* g1  = (const float*)d_in[9];
  const float* be1 = (const float*)d_in[10];
  const float* Wfc = (const float*)d_in[11];
  const float* bfc = (const float*)d_in[12];
  float* out = (float*)d_out;

  const long long N = in_sizes[0] / 128;
  const long long E = in_sizes[1] / 2;

  char* ws = (char*)d_ws;
  size_t off = 0;
  auto carve = [&](size_t bytes) -> float* {
    char* p = ws + off;
    off += (bytes + 255) & ~(size_t)255;
    return (float*)p;
  };
  float* dinv   = carve((size_t)N * 4);        // deg -> rsqrt(deg) in place
  float* hbuf   = carve((size_t)N * 128 * 4);  // transform output
  float* agg    = carve((size_t)N * 128 * 4);  // scatter target / bn in-place
  float* stats  = carve(256 * 4);              // [sum|sumsq]
  float* pooled = carve(64 * 128 * 4);
  float* cnt    = carve(64 * 4);
  (void)ws_size; (void)n_in; (void)out_size;

  const dim3 blk(256);
  const long long nBlkN    = (N + 255) / 256;
  const long long nBlkE    = (E + 255) / 256;
  const long long gemmBlk  = (N + 127) / 128;
  const long long edgeBlk  = ((E + N) * 32 + 255) / 256;
  const long long applyBlk = (N * 32 + 255) / 256;

  // GCN symmetric normalization on A + I
  k_init_deg<<<nBlkN, blk, 0, stream>>>(dinv, N);
  k_deg_edges<<<nBlkE, blk, 0, stream>>>(ei, dinv, E);
  k_deg_to_dinv<<<nBlkN, blk, 0, stream>>>(dinv, N);

  // ---- layer 1 ----
  k_gcn_gemm128<<<gemmBlk, blk, 0, stream>>>(x, W0, hbuf, N);
  hipMemsetAsync(agg, 0, (size_t)N * 128 * 4, stream);
  k_edge_agg<<<edgeBlk, blk, 0, stream>>>(hbuf, ei, dinv, agg, E, N);
  hipMemsetAsync(stats, 0, 256 * 4, stream);
  k_bn_stats<<<512, blk, 0, stream>>>(agg, stats, N);
  k_bn_apply<<<applyBlk, blk, 0, stream>>>(agg, stats, g0, be0, N);

  // ---- layer 2 ----
  k_gcn_gemm128<<<gemmBlk, blk, 0, stream>>>(agg, W1, hbuf, N);
  hipMemsetAsync(agg, 0, (size_t)N * 128 * 4, stream);  // gemm consumed agg already
  k_edge_agg<<<edgeBlk, blk, 0, stream>>>(hbuf, ei, dinv, agg, E, N);
  hipMemsetAsync(stats, 0, 256 * 4, stream);
  k_bn_stats<<<512, blk, 0, stream>>>(agg, stats, N);
  k_bn_apply<<<applyBlk, blk, 0, stream>>>(agg, stats, g1, be1, N);

  // ---- pool + FC ----
  hipMemsetAsync(pooled, 0, 64 * 128 * 4, stream);
  hipMemsetAsync(cnt, 0, 64 * 4, stream);
  k_cnt<<<nBlkN, blk, 0, stream>>>(batch, cnt, N);
  k_pool<<<applyBlk, blk, 0, stream>>>(agg, batch, pooled, N);
  k_pool_fc<<<1, blk, 0, stream>>>(pooled, cnt, Wfc, bfc, out);
}